// SparseMatrixEquivariantLayerBlock_56186762166418
// MI455X (gfx1250) — compile-verified
//
#include <hip/hip_runtime.h>

#define N_NODE 100000
#define D      64
#define PAD    68            // LDS row pitch in floats: bank = (4*idx + k) % 64 -> conflict-free b64

typedef float v2f __attribute__((ext_vector_type(2)));
typedef float v8f __attribute__((ext_vector_type(8)));

// D(16x16,f32) = A(16x4,f32) * B(4x16,f32) + C   -- exact fp32 matrix pipe
__device__ __forceinline__ v8f wmma4(v2f a, v2f b, v8f c) {
    return __builtin_amdgcn_wmma_f32_16x16x4_f32(false, a, false, b, (short)0, c, false, false);
}

__global__ void k_zero(float* __restrict__ p, int count) {
    int i = blockIdx.x * blockDim.x + threadIdx.x;
    int stride = gridDim.x * blockDim.x;
    for (; i < count; i += stride) p[i] = 0.0f;
}

// Scatter-add: r += vals by rows, c += vals by cols, dv += diag vals; s,t global sums.
__global__ __launch_bounds__(256) void k_scatter(
    const float* __restrict__ vals, const int* __restrict__ rows, const int* __restrict__ cols,
    float* __restrict__ r, float* __restrict__ c, float* __restrict__ dv,
    float* __restrict__ s, float* __restrict__ t, int nnz) {
    __shared__ float ls[D], lt[D];
    int tid = threadIdx.x;
    if (tid < D) { ls[tid] = 0.0f; lt[tid] = 0.0f; }
    __syncthreads();

    int g = blockIdx.x * 256 + tid;
    int e  = g >> 3;
    int f0 = (g & 7) * 8;
    if (e < nnz) {
        const float4* v4 = (const float4*)(vals + (size_t)e * D + f0);
        float4 va = v4[0], vb = v4[1];
        float v[8] = {va.x, va.y, va.z, va.w, vb.x, vb.y, vb.z, vb.w};
        int rr = rows[e], cc = cols[e];
        float* rp = r + (size_t)rr * D + f0;
        float* cp = c + (size_t)cc * D + f0;
        #pragma unroll
        for (int j = 0; j < 8; j++) {
            atomicAdd(rp + j, v[j]);
            atomicAdd(cp + j, v[j]);
            atomicAdd(&ls[f0 + j], v[j]);
        }
        if (rr == cc) {
            float* dp = dv + (size_t)rr * D + f0;
            #pragma unroll
            for (int j = 0; j < 8; j++) {
                atomicAdd(dp + j, v[j]);
                atomicAdd(&lt[f0 + j], v[j]);
            }
        }
    }
    __syncthreads();
    if (tid < D) { atomicAdd(s + tid, ls[tid]); atomicAdd(t + tid, lt[tid]); }
}

// glob = s@W8 + t@W9 ; glob_diag = s@W13 + t@W14 ; bsum = sum(bias)
__global__ void k_glob(const float* __restrict__ W, const float* __restrict__ bias,
                       const float* __restrict__ s, const float* __restrict__ t,
                       float* __restrict__ glob, float* __restrict__ gdiag,
                       float* __restrict__ bsum) {
    int j = threadIdx.x;
    float g = 0.0f, gd = 0.0f;
    for (int k = 0; k < D; k++) {
        float sk = s[k], tk = t[k];
        g  += sk * W[8  * 4096 + k * D + j] + tk * W[9  * 4096 + k * D + j];
        gd += sk * W[13 * 4096 + k * D + j] + tk * W[14 * 4096 + k * D + j];
    }
    glob[j] = g;
    gdiag[j] = gd;
    if (j == 0) { float b = 0.0f; for (int i = 0; i < 15; i++) b += bias[i]; *bsum = b; }
}

// out = r@W[wr] + c@W[wc] + dv@W[wd], shape [N_NODE, 64]. One wave per 16-row tile.
__global__ __launch_bounds__(256) void k_nodegemm(
    const float* __restrict__ r, const float* __restrict__ c, const float* __restrict__ dv,
    const float* __restrict__ W, int wr, int wc, int wd, float* __restrict__ out) {
    __shared__ float wlds[3 * D * PAD];    // 3 transposed 64x64 weights, pad-68 rows (52.2 KB)
    int tid = threadIdx.x;
    for (int idx = tid; idx < 3 * 4096; idx += 256) {
        int w = idx >> 12, rem = idx & 4095, k = rem >> 6, n = rem & 63;
        int wsel = (w == 0) ? wr : ((w == 1) ? wc : wd);
        wlds[w * (D * PAD) + n * PAD + k] = W[(size_t)wsel * 4096 + rem];   // Wt[n][k] = W[k][n]
    }
    __syncthreads();

    int wave = tid >> 5, lane = tid & 31;
    int tile = blockIdx.x * 8 + wave;
    if (tile >= N_NODE / 16) return;

    int m = lane & 15, half = lane >> 4;
    int arow = tile * 16 + m;

    // A fragments (ISA f32 16x4 layout: lanes 0-15 -> K=4kk..4kk+1, lanes 16-31 -> K=4kk+2..+3)
    v2f ar[16], ac[16], ad[16];
    #pragma unroll
    for (int kk = 0; kk < 16; kk++) {
        int off = arow * D + 4 * kk + 2 * half;
        ar[kk] = *(const v2f*)(r  + off);
        ac[kk] = *(const v2f*)(c  + off);
        ad[kk] = *(const v2f*)(dv + off);
    }
    const float* w0 = wlds;
    const float* w1 = wlds + D * PAD;
    const float* w2 = wlds + 2 * D * PAD;

    for (int nn = 0; nn < 4; nn++) {
        v8f acc = {};
        int bn = (m + 16 * nn) * PAD;      // B column n = (lane&15) + 16*nn
        #pragma unroll
        for (int kk = 0; kk < 16; kk++) {
            int k0 = 4 * kk + 2 * half;    // B layout: VGPR0=K=k0, VGPR1=K=k0+1 per lane half
            acc = wmma4(ar[kk], *(const v2f*)(w0 + bn + k0), acc);
            acc = wmma4(ac[kk], *(const v2f*)(w1 + bn + k0), acc);
            acc = wmma4(ad[kk], *(const v2f*)(w2 + bn + k0), acc);
        }
        int ncol = 16 * nn + m;
        #pragma unroll
        for (int v = 0; v < 8; v++) {      // D layout: VGPR v -> row v+8*half
            int row = tile * 16 + v + 8 * half;
            out[(size_t)row * D + ncol] = acc[v];
        }
    }
}

// Y = vals@W0 + vals[perm]@W1 + row_d[rows] + col_d[cols] + glob + diag*(diag_d[rows]+gdiag) + bsum
__global__ __launch_bounds__(128) void k_edge(
    const float* __restrict__ vals, const int* __restrict__ rows, const int* __restrict__ cols,
    const int* __restrict__ perm, const float* __restrict__ W,
    const float* __restrict__ row_d, const float* __restrict__ col_d, const float* __restrict__ diag_d,
    const float* __restrict__ glob, const float* __restrict__ gdiag, const float* __restrict__ bsum,
    float* __restrict__ out, int nnz) {
    __shared__ float wlds[2 * D * PAD];        // W0^T, W1^T (34.8 KB)
    __shared__ float atile[4][16 * PAD];       // one swizzle-staging tile per wave (17.4 KB)
    int tid = threadIdx.x;
    for (int idx = tid; idx < 2 * 4096; idx += 128) {
        int w = idx >> 12, rem = idx & 4095, k = rem >> 6, n = rem & 63;
        wlds[w * (D * PAD) + n * PAD + k] = W[(size_t)w * 4096 + rem];
    }
    __syncthreads();

    int wave = tid >> 5, lane = tid & 31;
    int ntiles = (nnz + 15) >> 4;
    int tile = blockIdx.x * 4 + wave;
    if (tile >= ntiles) return;

    int e0 = tile * 16;
    float* at = atile[wave];
    int m = lane & 15, half = lane >> 4;

    // Stage vals tile coalesced (256B per row), then read swizzled A fragments.
    #pragma unroll 4
    for (int row = 0; row < 16; row++) {
        int e = min(e0 + row, nnz - 1);
        *(v2f*)(at + row * PAD + 2 * lane) = *(const v2f*)(vals + (size_t)e * D + 2 * lane);
    }
    v2f a0[16];
    #pragma unroll
    for (int kk = 0; kk < 16; kk++)
        a0[kk] = *(const v2f*)(at + m * PAD + 4 * kk + 2 * half);

    // Gather permuted rows into the same wave-local buffer (DS ops are in-order per wave).
    #pragma unroll 4
    for (int row = 0; row < 16; row++) {
        int e = min(e0 + row, nnz - 1);
        int pe = perm[e];
        *(v2f*)(at + row * PAD + 2 * lane) = *(const v2f*)(vals + (size_t)pe * D + 2 * lane);
    }
    v2f a1[16];
    #pragma unroll
    for (int kk = 0; kk < 16; kk++)
        a1[kk] = *(const v2f*)(at + m * PAD + 4 * kk + 2 * half);

    int re[8], ce[8];
    #pragma unroll
    for (int v = 0; v < 8; v++) {
        int e = min(e0 + v + 8 * half, nnz - 1);
        re[v] = rows[e];
        ce[v] = cols[e];
    }
    float bs = *bsum;

    for (int nn = 0; nn < 4; nn++) {
        v8f acc = {};
        int bn = (m + 16 * nn) * PAD;
        #pragma unroll
        for (int kk = 0; kk < 16; kk++) {
            int k0 = 4 * kk + 2 * half;
            acc = wmma4(a0[kk], *(const v2f*)(wlds + bn + k0), acc);
            acc = wmma4(a1[kk], *(const v2f*)(wlds + D * PAD + bn + k0), acc);
        }
        int ncol = 16 * nn + m;
        float gn = glob[ncol], gdn = gdiag[ncol];
        #pragma unroll
        for (int v = 0; v < 8; v++) {
            int e = e0 + v + 8 * half;
            if (e < nnz) {
                float y = acc[v] + row_d[(size_t)re[v] * D + ncol]
                                 + col_d[(size_t)ce[v] * D + ncol] + gn + bs;
                if (re[v] == ce[v]) y += diag_d[(size_t)re[v] * D + ncol] + gdn;
                out[(size_t)e * D + ncol] = y;
            }
        }
    }
}

extern "C" void kernel_launch(void* const* d_in, const int* in_sizes, int n_in,
                              void* d_out, int out_size, void* d_ws, size_t ws_size,
                              hipStream_t stream) {
    const float* vals = (const float*)d_in[0];
    const int*   rows = (const int*)d_in[1];
    const int*   cols = (const int*)d_in[2];
    const int*   perm = (const int*)d_in[3];
    const float* W    = (const float*)d_in[4];
    const float* bias = (const float*)d_in[5];
    int nnz = in_sizes[1];

    float* ws = (float*)d_ws;
    size_t nd = (size_t)N_NODE * D;
    float* r      = ws;
    float* c      = r + nd;
    float* dv     = c + nd;
    float* row_d  = dv + nd;
    float* col_d  = row_d + nd;
    float* diag_d = col_d + nd;
    float* smalls = diag_d + nd;          // s[64] t[64] glob[64] gdiag[64] bsum[1]
    float* s = smalls, *t = smalls + 64, *glob = smalls + 128, *gdiag = smalls + 192, *bsum = smalls + 256;

    k_zero<<<2048, 256, 0, stream>>>(r, (int)(3 * nd));     // r, c, dv are contiguous
    k_zero<<<1, 256, 0, stream>>>(smalls, 128);             // s, t

    int sgrid = (nnz * 8 + 255) / 256;
    k_scatter<<<sgrid, 256, 0, stream>>>(vals, rows, cols, r, c, dv, s, t, nnz);

    k_glob<<<1, 64, 0, stream>>>(W, bias, s, t, glob, gdiag, bsum);

    int ngrid = (N_NODE / 16 + 7) / 8;
    k_nodegemm<<<ngrid, 256, 0, stream>>>(r, c, dv, W, 2, 4, 6, row_d);    // row_d
    k_nodegemm<<<ngrid, 256, 0, stream>>>(r, c, dv, W, 3, 5, 7, col_d);    // col_d
    k_nodegemm<<<ngrid, 256, 0, stream>>>(r, c, dv, W, 11, 12, 10, diag_d); // diag_d = r@W11+c@W12+dv@W10

    int ntiles = (nnz + 15) / 16;
    int egrid = (ntiles + 3) / 4;
    k_edge<<<egrid, 128, 0, stream>>>(vals, rows, cols, perm, W,
                                      row_d, col_d, diag_d, glob, gdiag, bsum,
                                      (float*)d_out, nnz);
}